// ROITokenCompression_3753801417563
// MI455X (gfx1250) — compile-verified
//
#include <hip/hip_runtime.h>
#include <hip/hip_bf16.h>
#include <math.h>

// ---------------- problem constants ----------------
constexpr int Bc = 4, Tc = 16, Nc = 1024, Dc = 768;
constexpr int Kc = 64, Gc = 4, Hc = 8, HDc = 96;
constexpr int Fc = Dc / 2;          // 384
constexpr int BT = Bc * Tc;         // 64
constexpr int Mrows = BT * Nc;      // 65536 total token rows
constexpr float SCALE = 0.10206207261596577f; // 1/sqrt(96)

typedef float v2f __attribute__((ext_vector_type(2)));
typedef float v8f __attribute__((ext_vector_type(8)));

// D = A(16x4) * B(4x16) + C  (fp32 WMMA, wave32)
__device__ __forceinline__ v8f wmma4(v2f a, v2f b, v8f c) {
  return __builtin_amdgcn_wmma_f32_16x16x4_f32(
      /*neg_a=*/false, a, /*neg_b=*/false, b,
      /*c_mod=*/(short)0, c, /*reuse_a=*/false, /*reuse_b=*/false);
}

__device__ __forceinline__ float gelu_exact(float x) {
  return 0.5f * x * (1.0f + erff(x * 0.70710678118654752f));
}

// ================= kernel 0: q, A, qb precompute (constant over b,t) =========
__global__ __launch_bounds__(256)
void precompute_kernel(const float* __restrict__ gt, const float* __restrict__ wq,
                       const float* __restrict__ bq, const float* __restrict__ wk,
                       const float* __restrict__ bk,
                       float* __restrict__ Amat, float* __restrict__ qb) {
  __shared__ float q[Gc * Dc]; // 12 KB
  for (int o = threadIdx.x; o < Gc * Dc; o += 256) {
    int g = o / Dc, dd = o % Dc;
    float s = bq[dd];
    const float* gp = gt + (size_t)g * Dc;
    for (int e = 0; e < Dc; ++e) s += gp[e] * wq[(size_t)e * Dc + dd];
    q[o] = s;
  }
  __syncthreads();
  // A[(h*G+g), e] = sum_{d<HD} q[g, h*HD+d] * wk[e, h*HD+d]
  for (int o = threadIdx.x; o < Hc * Gc * Dc; o += 256) {
    int hg = o / Dc, e = o % Dc;
    int h = hg / Gc, g = hg % Gc;
    const float* qp = q + g * Dc + h * HDc;
    const float* wkp = wk + (size_t)e * Dc + h * HDc;
    float s = 0.f;
    #pragma unroll 8
    for (int d = 0; d < HDc; ++d) s += qp[d] * wkp[d];
    Amat[o] = s;
  }
  if (threadIdx.x < Hc * Gc) {
    int h = threadIdx.x / Gc, g = threadIdx.x % Gc;
    float s = 0.f;
    for (int d = 0; d < HDc; ++d) s += q[g * Dc + h * HDc + d] * bk[h * HDc + d];
    qb[threadIdx.x] = s;
  }
}

// ================= kernel 1: scorer GEMM + GELU + w2 dot + ROI bias ==========
// Each wave owns one 16-row tile of the flattened [65536 x 768] token matrix.
__global__ __launch_bounds__(256)
void scorer_kernel(const float* __restrict__ tokens, const int* __restrict__ roi,
                   const float* __restrict__ w1, const float* __restrict__ b1,
                   const float* __restrict__ w2, const float* __restrict__ b2,
                   float* __restrict__ scores) {
  const int wave = threadIdx.x >> 5;
  const int lane = threadIdx.x & 31;
  const int tile = blockIdx.x * 8 + wave;
  const int row0 = tile * 16;
  const int mrow = row0 + (lane & 15);
  const int khalf = (lane >> 4) * 2;          // A frag: lanes 0-15 K=0,1; 16-31 K=2,3
  const float* arow = tokens + (size_t)mrow * Dc;

  float acc[8];
  #pragma unroll
  for (int j = 0; j < 8; ++j) acc[j] = 0.f;

  for (int fg = 0; fg < 4; ++fg) {            // 4 groups of 6 F-tiles (96 cols)
    v8f C[6];
    #pragma unroll
    for (int ft = 0; ft < 6; ++ft) C[ft] = {};
    for (int k4 = 0; k4 < Dc / 4; ++k4) {
      const int kb = k4 * 4 + khalf;
      if ((k4 & 63) == 0 && kb + 256 < Dc)    // stream-prefetch the A row
        __builtin_prefetch(arow + kb + 256, 0, 0);
      v2f a = *(const v2f*)(arow + kb);       // contiguous b64 load
      #pragma unroll
      for (int ft = 0; ft < 6; ++ft) {
        const int col = (fg * 6 + ft) * 16 + (lane & 15);
        v2f b;
        b.x = w1[(size_t)kb * Fc + col];
        b.y = w1[(size_t)(kb + 1) * Fc + col];
        C[ft] = wmma4(a, b, C[ft]);
      }
    }
    #pragma unroll
    for (int ft = 0; ft < 6; ++ft) {
      const int col = (fg * 6 + ft) * 16 + (lane & 15);
      const float w2c = w2[col], b1c = b1[col];
      #pragma unroll
      for (int j = 0; j < 8; ++j) {
        float hx = gelu_exact(C[ft][j] + b1c);
        acc[j] += hx * w2c;
      }
    }
  }
  // reduce the 16 columns held across each 16-lane half
  #pragma unroll
  for (int j = 0; j < 8; ++j) {
    #pragma unroll
    for (int off = 1; off < 16; off <<= 1)
      acc[j] += __shfl_xor(acc[j], off, 32);
  }
  const float b2v = b2[0];
  if ((lane & 15) == 0) {
    const int base = row0 + ((lane >> 4) ? 8 : 0);
    #pragma unroll
    for (int j = 0; j < 8; ++j) {
      const int r = base + j;
      const float rb = roi[r] ? 2.0f : 1.0f;  // ROI_WEIGHT=2 -> mask*(2-1)+1
      scores[r] = (acc[j] + b2v) * rb;
    }
  }
}

// ================= kernel 2: exact top-K by rank (lax.top_k tie order) =======
__global__ __launch_bounds__(256)
void topk_kernel(const float* __restrict__ scores, int* __restrict__ idx_ws,
                 float* __restrict__ out_indices) {
  __shared__ float s[Nc];
  __shared__ int sel[Kc];
  const int bt = blockIdx.x;
  const float* sp = scores + (size_t)bt * Nc;
  for (int i = threadIdx.x; i < Nc; i += 256) s[i] = sp[i];
  __syncthreads();
  for (int i = threadIdx.x; i < Nc; i += 256) {
    const float v = s[i];
    int rank = 0;
    for (int j = 0; j < Nc; ++j) {
      const float u = s[j];
      rank += (u > v) || (u == v && j < i);
    }
    if (rank < Kc) sel[rank] = i;
  }
  __syncthreads();
  if (threadIdx.x < Kc) {
    const int i = sel[threadIdx.x];
    idx_ws[bt * Kc + threadIdx.x] = i;
    out_indices[bt * Kc + threadIdx.x] = (float)i;
  }
}

// ================= kernel 3: gather selected rows (float4, dual write) =======
__global__ __launch_bounds__(192)
void gather_kernel(const float* __restrict__ tokens, const int* __restrict__ idx_ws,
                   float* __restrict__ out_selected, float* __restrict__ out_comp) {
  const int btk = blockIdx.x;
  const int bt = btk / Kc, kk = btk % Kc;
  const int n = idx_ws[btk];
  const float4* src = (const float4*)(tokens + ((size_t)bt * Nc + n) * Dc);
  float4* ds = (float4*)(out_selected + (size_t)btk * Dc);
  float4* dc = (float4*)(out_comp + ((size_t)bt * (Kc + Gc) + kk) * Dc);
  const float4 v = src[threadIdx.x];
  ds[threadIdx.x] = v;
  dc[threadIdx.x] = v;
}

// ================= kernel 4: logits = tokens @ A^T  (WMMA GEMM) ==============
// Output layout transposed to L[bt][hg][n] so softmax/usum are coalesced.
__global__ __launch_bounds__(256)
void logits_kernel(const float* __restrict__ tokens, const float* __restrict__ Amat,
                   const float* __restrict__ qb, float* __restrict__ logit) {
  const int wave = threadIdx.x >> 5, lane = threadIdx.x & 31;
  const int mtile = blockIdx.x * 4 + (wave >> 1);
  const int ncol = (wave & 1) * 16 + (lane & 15);     // 0..31 (h*G+g)
  const int mrow = mtile * 16 + (lane & 15);
  const int khalf = (lane >> 4) * 2;
  const float* arow = tokens + (size_t)mrow * Dc;
  const float* brow = Amat + (size_t)ncol * Dc;       // B[k][n] = A[n][k] (contig pair)
  v8f C = {};
  for (int k4 = 0; k4 < Dc / 4; ++k4) {
    const int kb = k4 * 4 + khalf;
    if ((k4 & 63) == 0 && kb + 256 < Dc)
      __builtin_prefetch(arow + kb + 256, 0, 0);
    v2f a = *(const v2f*)(arow + kb);
    v2f b = *(const v2f*)(brow + kb);
    C = wmma4(a, b, C);
  }
  const float qbv = qb[ncol];
  // rows rbase..rbase+7 are consecutive token indices n -> two b128 stores
  const int grow = mtile * 16 + ((lane >> 4) ? 8 : 0);
  const int btq = grow >> 10, n0 = grow & (Nc - 1);
  float* dst = logit + (((size_t)btq * 32 + ncol) << 10) + n0;
  float4 lo, hi;
  lo.x = (C[0] + qbv) * SCALE; lo.y = (C[1] + qbv) * SCALE;
  lo.z = (C[2] + qbv) * SCALE; lo.w = (C[3] + qbv) * SCALE;
  hi.x = (C[4] + qbv) * SCALE; hi.y = (C[5] + qbv) * SCALE;
  hi.z = (C[6] + qbv) * SCALE; hi.w = (C[7] + qbv) * SCALE;
  *(float4*)dst = lo;
  *(float4*)(dst + 4) = hi;
}

// ================= kernel 5: softmax over N, one wave per (bt,hg) row ========
__global__ __launch_bounds__(256)
void softmax_kernel(float* __restrict__ logit) {
  const int wave = threadIdx.x >> 5, lane = threadIdx.x & 31;
  const int rid = blockIdx.x * 8 + wave;              // 0..2047 = bt*32+hg
  float* row = logit + ((size_t)rid << 10);           // 1024 contiguous floats
  float vals[32];
  float mx = -INFINITY;
  #pragma unroll
  for (int i = 0; i < 32; ++i) {
    vals[i] = row[i * 32 + lane];                     // fully coalesced
    mx = fmaxf(mx, vals[i]);
  }
  #pragma unroll
  for (int off = 16; off >= 1; off >>= 1) mx = fmaxf(mx, __shfl_xor(mx, off, 32));
  float sum = 0.f;
  #pragma unroll
  for (int i = 0; i < 32; ++i) { vals[i] = __expf(vals[i] - mx); sum += vals[i]; }
  #pragma unroll
  for (int off = 16; off >= 1; off >>= 1) sum += __shfl_xor(sum, off, 32);
  const float inv = 1.f / sum;
  #pragma unroll
  for (int i = 0; i < 32; ++i) row[i * 32 + lane] = vals[i] * inv;
}

// ================= kernel 6: u = attn @ tokens  per (b,t)  (WMMA) ============
__global__ __launch_bounds__(256)
void usum_kernel(const float* __restrict__ attn, const float* __restrict__ tokens,
                 float* __restrict__ U) {
  const int wave = threadIdx.x >> 5, lane = threadIdx.x & 31;
  const int bt = blockIdx.x / 12;
  const int et = blockIdx.x % 12;                     // 64-wide E tile
  const int mi = wave >> 2;                           // 0..1 (hg tile)
  const int ei = wave & 3;                            // 0..3
  const int m = mi * 16 + (lane & 15);
  const int ecol = et * 64 + ei * 16 + (lane & 15);
  const int khalf = (lane >> 4) * 2;
  const float* arow = attn + (((size_t)bt * 32 + m) << 10); // A[m][k] contiguous
  const float* tb = tokens + (size_t)bt * Nc * Dc;
  v8f C = {};
  for (int k4 = 0; k4 < Nc / 4; ++k4) {
    const int kb = k4 * 4 + khalf;
    v2f a = *(const v2f*)(arow + kb);                 // contiguous b64 load
    v2f b; b.x = tb[(size_t)kb * Dc + ecol];    b.y = tb[(size_t)(kb + 1) * Dc + ecol];
    C = wmma4(a, b, C);
  }
  const int rbase = mi * 16 + ((lane >> 4) ? 8 : 0);
  float* ub = U + (size_t)bt * 32 * Dc;
  #pragma unroll
  for (int j = 0; j < 8; ++j)
    ub[(size_t)(rbase + j) * Dc + ecol] = C[j];
}

// ================= kernel 7: ctx = u @ wv_head + bv ; out = ctx @ wo + bo ====
__global__ __launch_bounds__(256)
void attnout_kernel(const float* __restrict__ U, const float* __restrict__ wv,
                    const float* __restrict__ bv, const float* __restrict__ wo,
                    const float* __restrict__ bo, float* __restrict__ out_global,
                    float* __restrict__ out_comp) {
  __shared__ float ctx[Gc * Dc]; // 12 KB
  const int bt = blockIdx.x;
  const float* ub = U + (size_t)bt * 32 * Dc;
  for (int o = threadIdx.x; o < Gc * Dc; o += 256) {
    const int g = o / Dc, dd = o % Dc;
    const int h = dd / HDc;
    const float* up = ub + (size_t)(h * Gc + g) * Dc;
    float s = bv[dd];
    for (int e = 0; e < Dc; ++e) s += up[e] * wv[(size_t)e * Dc + dd];
    ctx[o] = s;
  }
  __syncthreads();
  for (int o = threadIdx.x; o < Gc * Dc; o += 256) {
    const int g = o / Dc, d = o % Dc;
    const float* cg = ctx + g * Dc;
    float s = bo[d];
    for (int dd = 0; dd < Dc; ++dd) s += cg[dd] * wo[(size_t)dd * Dc + d];
    out_global[(size_t)bt * Gc * Dc + o] = s;
    out_comp[((size_t)bt * (Kc + Gc) + Kc + g) * Dc + d] = s;
  }
}

// ============================== launcher =====================================
extern "C" void kernel_launch(void* const* d_in, const int* in_sizes, int n_in,
                              void* d_out, int out_size, void* d_ws, size_t ws_size,
                              hipStream_t stream) {
  const float* tokens = (const float*)d_in[0];
  const int*   roi    = (const int*)d_in[1];
  const float* w1     = (const float*)d_in[2];
  const float* b1     = (const float*)d_in[3];
  const float* w2     = (const float*)d_in[4];
  const float* b2     = (const float*)d_in[5];
  const float* gt     = (const float*)d_in[6];
  const float* wq     = (const float*)d_in[7];
  const float* bq     = (const float*)d_in[8];
  const float* wk     = (const float*)d_in[9];
  const float* bk     = (const float*)d_in[10];
  const float* wv     = (const float*)d_in[11];
  const float* bv     = (const float*)d_in[12];
  const float* wo     = (const float*)d_in[13];
  const float* bo     = (const float*)d_in[14];

  float* out_compressed = (float*)d_out;                           // BT*(K+G)*D
  float* out_selected   = out_compressed + (size_t)BT * (Kc + Gc) * Dc;
  float* out_global     = out_selected + (size_t)BT * Kc * Dc;
  float* out_indices    = out_global + (size_t)BT * Gc * Dc;
  float* out_scores     = out_indices + (size_t)BT * Kc;

  float* wsf   = (float*)d_ws;
  float* Amat  = wsf;                              // 32*768
  float* qb    = Amat + Hc * Gc * Dc;              // 32
  float* logit = qb + 32;                          // 64*32*1024 ([bt][hg][n])
  float* U     = logit + (size_t)Mrows * 32;       // 64*32*768
  int*   idxws = (int*)(U + (size_t)BT * 32 * Dc); // 64*64 ints

  precompute_kernel<<<1, 256, 0, stream>>>(gt, wq, bq, wk, bk, Amat, qb);
  scorer_kernel<<<Mrows / 16 / 8, 256, 0, stream>>>(tokens, roi, w1, b1, w2, b2,
                                                    out_scores);
  topk_kernel<<<BT, 256, 0, stream>>>(out_scores, idxws, out_indices);
  gather_kernel<<<BT * Kc, 192, 0, stream>>>(tokens, idxws, out_selected,
                                             out_compressed);
  logits_kernel<<<Mrows / 16 / 4, 256, 0, stream>>>(tokens, Amat, qb, logit);
  softmax_kernel<<<BT * 32 / 8, 256, 0, stream>>>(logit);
  usum_kernel<<<BT * 12, 256, 0, stream>>>(logit, tokens, U);
  attnout_kernel<<<BT, 256, 0, stream>>>(U, wv, bv, wo, bo, out_global,
                                         out_compressed);
}